// HeteroMPNNPredictor_57303453663601
// MI455X (gfx1250) — compile-verified
//
#include <hip/hip_runtime.h>
#include <hip/hip_bf16.h>

// ---------------------------------------------------------------------------
// HeteroMPNNPredictor on MI455X (gfx1250), fp32 end-to-end.
// GEMMs via V_WMMA_F32_16X16X4_F32 (fp32 matrix path, no precision loss).
// Message passing: precomputed per-etype inverse degrees + f32 global atomics.
// ---------------------------------------------------------------------------

typedef __attribute__((ext_vector_type(2))) float v2f;
typedef __attribute__((ext_vector_type(8))) float v8f;

#define N_CFG 20000
#define N_AST 100000
#define N_PT  50
#define N_FT  10
#define NTOT  120060L   // N_CFG + N_AST + N_PT + N_FT

static const int  NTYPE[4]   = {N_CFG, N_AST, N_PT, N_FT};
static const long TOFF[4]    = {0, 20000, 120000, 120050};
static const int  ET_S[8]    = {0, 1, 1, 0, 2, 0, 3, 0};   // src type per etype
static const int  ET_D[8]    = {0, 1, 0, 1, 0, 2, 0, 3};   // dst type per etype
static const int  ET_E[8]    = {80000, 300000, 100000, 100000, 100000, 100000, 40000, 40000};
static const long INV_OFF[8] = {0, 20000, 120000, 140000, 240000, 260000, 260050, 280050};
#define INV_TOTAL 280060L

// ---------------------------------------------------------------------------
// utility kernels
// ---------------------------------------------------------------------------
__global__ void k_zero(float* __restrict__ p, long n) {
  long i = (long)blockIdx.x * blockDim.x + threadIdx.x;
  if (i < n) p[i] = 0.0f;
}

__global__ void k_broadcast(const float* __restrict__ emb, float* __restrict__ out,
                            int n, int w) {
  long i = (long)blockIdx.x * blockDim.x + threadIdx.x;
  if (i < (long)n * w) out[i] = emb[i & (w - 1)];   // w == 128 (power of two)
}

__global__ void k_count(const int* __restrict__ dst, int E, float* __restrict__ cnt) {
  int e = blockIdx.x * blockDim.x + threadIdx.x;
  if (e < E) unsafeAtomicAdd(cnt + dst[e], 1.0f);
}

__global__ void k_invert(float* __restrict__ cnt, long n) {
  long i = (long)blockIdx.x * blockDim.x + threadIdx.x;
  if (i < n) cnt[i] = 1.0f / fmaxf(cnt[i], 1.0f);
}

// concat helper: dst[r*wdst + col0 + c] = src[r*wsrc + c]
__global__ void k_copycols(const float* __restrict__ src, float* __restrict__ dst,
                           long rows, int wsrc, int wdst, int col0) {
  long i = (long)blockIdx.x * blockDim.x + threadIdx.x;
  if (i >= rows * (long)wsrc) return;
  long r = i / wsrc;
  int  c = (int)(i - r * wsrc);
  dst[r * (long)wdst + col0 + c] = src[i];
}

// ---------------------------------------------------------------------------
// fp32 WMMA GEMM:  C[r][col0 + n] = A[M x K] @ B[K x N] + bias[n]
// block = 256 threads = 8 waves arranged 4(M) x 2(N); block tile 64 x 64;
// each wave computes a 16(M) x 32(N) tile: two accumulators sharing one A
// fragment -> 2 independent WMMA chains per K-step for latency hiding.
// K multiple of 4, N multiple of 64 (guaranteed by this model).
//
// V_WMMA_F32_16X16X4_F32 layouts (ISA 7.12.2):
//   A 16x4 : lane&15 = M row; lane>=16 -> K+2; vgpr0 = K, vgpr1 = K+1
//   B 4x16 : lane&15 = N col; lane>=16 -> K+2; vgpr0 = K, vgpr1 = K+1
//   C 16x16: vgpr j = row j + 8*(lane>>4); lane&15 = N col
//
// Tail handling: A row pointer is clamped to M-1 (memory-safe); the garbage
// it feeds only reaches C rows r >= M, which the store mask drops. The K-loop
// is therefore completely branchless (pure loads + 2x WMMA + pointer adds).
// ---------------------------------------------------------------------------
__global__ __launch_bounds__(256)
void k_gemm_wmma(const float* __restrict__ A, const float* __restrict__ B,
                 const float* __restrict__ bias, float* __restrict__ C,
                 int M, int K, int N, int ldc, int col0) {
  int wave = threadIdx.x >> 5;
  int lane = threadIdx.x & 31;
  int m0 = (blockIdx.x * 4 + (wave & 3)) * 16;
  int n0 = (blockIdx.y * 2 + (wave >> 2)) * 32;   // always < N (N % 64 == 0)
  if (m0 >= M) return;                            // wave-uniform: EXEC all-1s

  int half = lane >> 4;                           // 0: K,K+1   1: K+2,K+3
  int l15  = lane & 15;
  int arow = m0 + l15;
  int arow_c = (arow < M) ? arow : (M - 1);       // clamp: safe, masked at store
  const float* Ap = A + (long)arow_c * K + half * 2;
  const float* Bp = B + (long)(half * 2) * N + n0 + l15;
  const long  Bstep = 4L * N;

  float bv0 = bias[n0 + l15];
  float bv1 = bias[n0 + 16 + l15];
  v8f c0, c1;
#pragma unroll
  for (int j = 0; j < 8; ++j) { c0[j] = bv0; c1[j] = bv1; }

  int iters = K >> 2;
#pragma unroll 2
  for (int it = 0; it < iters; ++it) {
    v2f a = *(const v2f*)Ap;                      // global_load_b64
    v2f b0, b1;
    b0[0] = Bp[0];  b0[1] = Bp[N];
    b1[0] = Bp[16]; b1[1] = Bp[N + 16];
    c0 = __builtin_amdgcn_wmma_f32_16x16x4_f32(false, a, false, b0,
                                               (short)0, c0, false, false);
    c1 = __builtin_amdgcn_wmma_f32_16x16x4_f32(false, a, false, b1,
                                               (short)0, c1, false, false);
    Ap += 4;
    Bp += Bstep;
  }

  int ro = half * 8;
#pragma unroll
  for (int j = 0; j < 8; ++j) {
    int r = m0 + j + ro;
    if (r < M) {
      float* Cr = C + (long)r * ldc + col0 + n0 + l15;
      Cr[0]  = c0[j];
      Cr[16] = c1[j];
    }
  }
}

// ---------------------------------------------------------------------------
// edge scatter with per-etype mean: out[dst] += Wh[src] * inv_deg[dst]
// one thread = 4 channels of one edge (float4 read, 4x f32 atomics into L2)
// ---------------------------------------------------------------------------
__global__ void k_scatter(const float* __restrict__ wh, const int* __restrict__ src,
                          const int* __restrict__ dst, int E,
                          const float* __restrict__ inv, float* __restrict__ out,
                          int dout) {
  int per = dout >> 2;
  long gid = (long)blockIdx.x * blockDim.x + threadIdx.x;
  if (gid >= (long)E * per) return;
  int e = (int)(gid / per);
  int c = (int)(gid - (long)e * per) * 4;
  int s = src[e], d = dst[e];
  float iv = inv[d];
  const float4 v = *(const float4*)(wh + (long)s * dout + c);
  float* o = out + (long)d * dout + c;
  unsafeAtomicAdd(o + 0, v.x * iv);
  unsafeAtomicAdd(o + 1, v.y * iv);
  unsafeAtomicAdd(o + 2, v.z * iv);
  unsafeAtomicAdd(o + 3, v.w * iv);
}

// ---------------------------------------------------------------------------
// decoder: logits = h @ W_dec + b_dec ; softmax over 2 classes
// ---------------------------------------------------------------------------
__global__ void k_decode(const float* __restrict__ h, const float* __restrict__ Wd,
                         const float* __restrict__ bd, float* __restrict__ logits,
                         float* __restrict__ soft, int n) {
  int i = blockIdx.x * blockDim.x + threadIdx.x;
  if (i >= n) return;
  const float* hp = h + (long)i * 128;
  float l0 = bd[0], l1 = bd[1];
#pragma unroll 8
  for (int k = 0; k < 128; ++k) {
    float v = hp[k];
    l0 += v * Wd[2 * k];
    l1 += v * Wd[2 * k + 1];
  }
  logits[2 * i]     = l0;
  logits[2 * i + 1] = l1;
  float m  = fmaxf(l0, l1);
  float e0 = expf(l0 - m), e1 = expf(l1 - m);
  float s  = e0 + e1;
  soft[2 * i]     = e0 / s;
  soft[2 * i + 1] = e1 / s;
}

// ---------------------------------------------------------------------------
// host-side helpers
// ---------------------------------------------------------------------------
static inline void launch_zero(float* p, long n, hipStream_t s) {
  k_zero<<<dim3((unsigned)((n + 255) / 256)), 256, 0, s>>>(p, n);
}

static inline void launch_gemm(const float* A, const float* B, const float* bias,
                               float* C, int M, int K, int N, int ldc, int col0,
                               hipStream_t s) {
  dim3 grid((M + 63) / 64, N / 64);
  k_gemm_wmma<<<grid, 256, 0, s>>>(A, B, bias, C, M, K, N, ldc, col0);
}

// one hetero-GCN layer: hin (slot width din) -> hout (slot width dout)
static void run_layer(const float* hin, int din, const float* W, const float* bias,
                      float* hout, int dout, const int* const* srcs,
                      const int* const* dsts, const float* invbase, float* wh,
                      hipStream_t s) {
  launch_zero(hout, NTOT * (long)dout, s);
  for (int i = 0; i < 8; ++i) {
    int st = ET_S[i], dt = ET_D[i];
    launch_gemm(hin + TOFF[st] * din, W + (long)i * din * dout,
                bias + (long)i * dout, wh, NTYPE[st], din, dout, dout, 0, s);
    long tot = (long)ET_E[i] * (dout >> 2);
    k_scatter<<<dim3((unsigned)((tot + 255) / 256)), 256, 0, s>>>(
        wh, srcs[i], dsts[i], ET_E[i], invbase + INV_OFF[i],
        hout + TOFF[dt] * dout, dout);
  }
}

extern "C" void kernel_launch(void* const* d_in, const int* in_sizes, int n_in,
                              void* d_out, int out_size, void* d_ws, size_t ws_size,
                              hipStream_t stream) {
  // ---- inputs (setup_inputs order) ----
  const float* cfg_label   = (const float*)d_in[0];
  const float* cfg_content = (const float*)d_in[1];
  const float* ast_label   = (const float*)d_in[2];
  const float* ast_content = (const float*)d_in[3];
  const int* srcs[8]; const int* dsts[8];
  for (int i = 0; i < 8; ++i) {
    srcs[i] = (const int*)d_in[4 + 2 * i];
    dsts[i] = (const int*)d_in[5 + 2 * i];
  }
  const float* W_cfg_lab = (const float*)d_in[20];
  const float* b_cfg_lab = (const float*)d_in[21];
  const float* W_cfg_con = (const float*)d_in[22];
  const float* b_cfg_con = (const float*)d_in[23];
  const float* W_ast_lab = (const float*)d_in[24];
  const float* b_ast_lab = (const float*)d_in[25];
  const float* W_ast_con = (const float*)d_in[26];
  const float* b_ast_con = (const float*)d_in[27];
  const float* ptest_emb = (const float*)d_in[28];
  const float* ftest_emb = (const float*)d_in[29];
  const float* W1 = (const float*)d_in[30]; const float* b1 = (const float*)d_in[31];
  const float* W2 = (const float*)d_in[32]; const float* b2 = (const float*)d_in[33];
  const float* W3 = (const float*)d_in[34]; const float* b3 = (const float*)d_in[35];
  const float* W4 = (const float*)d_in[36]; const float* b4 = (const float*)d_in[37];
  const float* W5 = (const float*)d_in[38]; const float* b5 = (const float*)d_in[39];
  const float* W_dec = (const float*)d_in[40];
  const float* b_dec = (const float*)d_in[41];
  float* out = (float*)d_out;

  // ---- workspace arena (floats); aliased lifetimes verified disjoint ----
  float* ws = (float*)d_ws;
  const long S128 = NTOT * 128L;              // 15,367,680
  float* h0    = ws;                          // w128  [A0]
  float* h1    = ws + S128;                   // w128  [A1]
  float* h2    = ws + 2 * S128;               // w128  [A2]
  float* cat12 = ws + 3 * S128;               // w256  [A3]
  float* h3    = h0;                          // w256  aliases A0+A1 (h0,h1 dead)
  float* h4    = cat12;                       // w256  aliases A3 (cat12 dead)
  float* cat34 = ws + 5 * S128;               // w512  [A5]
  float* h5    = h0;                          // w128  aliases A0 (h3 dead)
  float* wh    = ws + 9 * S128;               // 100000*256 = 25.6M floats max
  float* invd  = wh + 25600000L;              // 280,060 floats of 1/deg
  // total: 9*S128 + 25.6M + 0.28M ~= 164.2M floats ~= 657 MB

  // ---- per-etype inverse degrees (edge lists constant across layers) ----
  launch_zero(invd, INV_TOTAL, stream);
  for (int i = 0; i < 8; ++i)
    k_count<<<dim3((ET_E[i] + 255) / 256), 256, 0, stream>>>(
        dsts[i], ET_E[i], invd + INV_OFF[i]);
  k_invert<<<dim3((unsigned)((INV_TOTAL + 255) / 256)), 256, 0, stream>>>(invd, INV_TOTAL);

  // ---- encoders -> h0 (width 128; concat done via column offsets) ----
  launch_gemm(cfg_label,   W_cfg_lab, b_cfg_lab, h0 + TOFF[0] * 128, N_CFG, 80,  64, 128, 0,  stream);
  launch_gemm(cfg_content, W_cfg_con, b_cfg_con, h0 + TOFF[0] * 128, N_CFG, 300, 64, 128, 64, stream);
  launch_gemm(ast_label,   W_ast_lab, b_ast_lab, h0 + TOFF[1] * 128, N_AST, 80,  64, 128, 0,  stream);
  launch_gemm(ast_content, W_ast_con, b_ast_con, h0 + TOFF[1] * 128, N_AST, 300, 64, 128, 64, stream);
  k_broadcast<<<dim3((N_PT * 128 + 255) / 256), 256, 0, stream>>>(ptest_emb, h0 + TOFF[2] * 128, N_PT, 128);
  k_broadcast<<<dim3((N_FT * 128 + 255) / 256), 256, 0, stream>>>(ftest_emb, h0 + TOFF[3] * 128, N_FT, 128);

  // ---- GCN stack ----
  run_layer(h0, 128, W1, b1, h1, 128, srcs, dsts, invd, wh, stream);   // L1
  run_layer(h1, 128, W2, b2, h2, 128, srcs, dsts, invd, wh, stream);   // L2

  {  // cat12 = [h1 | h2]  (width 256)
    long n = NTOT * 128L;
    k_copycols<<<dim3((unsigned)((n + 255) / 256)), 256, 0, stream>>>(h1, cat12, NTOT, 128, 256, 0);
    k_copycols<<<dim3((unsigned)((n + 255) / 256)), 256, 0, stream>>>(h2, cat12, NTOT, 128, 256, 128);
  }

  run_layer(cat12, 256, W3, b3, h3, 256, srcs, dsts, invd, wh, stream); // L3
  run_layer(h3,    256, W4, b4, h4, 256, srcs, dsts, invd, wh, stream); // L4

  {  // cat34 = [h3 | h4]  (width 512)
    long n = NTOT * 256L;
    k_copycols<<<dim3((unsigned)((n + 255) / 256)), 256, 0, stream>>>(h3, cat34, NTOT, 256, 512, 0);
    k_copycols<<<dim3((unsigned)((n + 255) / 256)), 256, 0, stream>>>(h4, cat34, NTOT, 256, 512, 256);
  }

  run_layer(cat34, 512, W5, b5, h5, 128, srcs, dsts, invd, wh, stream); // L5

  // ---- decoder + softmax; output rows: [cfg_logits; cfg_soft; ast_logits; ast_soft] ----
  k_decode<<<dim3((N_CFG + 255) / 256), 256, 0, stream>>>(
      h5 + TOFF[0] * 128, W_dec, b_dec, out + 0, out + 2L * N_CFG, N_CFG);
  k_decode<<<dim3((N_AST + 255) / 256), 256, 0, stream>>>(
      h5 + TOFF[1] * 128, W_dec, b_dec, out + 4L * N_CFG, out + 4L * N_CFG + 2L * N_AST, N_AST);
}